// PoseurHandDecoder_24507083391016
// MI455X (gfx1250) — compile-verified
//
#include <hip/hip_runtime.h>
#include <math.h>

typedef __attribute__((ext_vector_type(2))) float v2f;
typedef __attribute__((ext_vector_type(4))) float v4f;
typedef __attribute__((ext_vector_type(8))) float v8f;

#define B_    32
#define D_    256
#define NH_   8
#define DH_   32
#define NLVL_ 4
#define NPTS_ 4
#define NQ_   42
#define NLAYERS_ 6
#define FF_   1024
#define S_TOT 5440   // 64*64 + 32*32 + 16*16 + 8*8

// ---------------------------------------------------------------------------
// Generic batched, strided GEMM:  C[b][m][n] = act( sum_k A[b][m][k]*B[k][n] + bias[n] )
// A element (m,k) at A + b*sAb + m*sAm + k*sAk  (supports the 1x1-conv layout)
// B element (k,n) at Bm + k*sBk + n             (supports column subviews)
// C element (m,n) at C + b*sCb + m*sCm + n
// Block: 256 threads = 8 waves. Block tile 32(M) x 128(N), K chunk 32.
// Each wave owns a 16x32 output strip = two v8f accumulators; one A fragment
// feeds two v_wmma_f32_16x16x4_f32 per K-substep (16 WMMA per wave per chunk).
// ---------------------------------------------------------------------------
__global__ __launch_bounds__(256) void gemm_wmma(
    const float* __restrict__ A, long long sAb, int sAm, int sAk,
    const float* __restrict__ Bm, int sBk,
    const float* __restrict__ bias,
    float* __restrict__ C, long long sCb, int sCm,
    int M, int N, int K, int act)
{
    __shared__ float As[32][36];    // 36*4 = 144B rows: 16B-aligned, bank-safe
    __shared__ float Bs[32][132];   // 132*4 = 528B rows: 16B-aligned, stride%64==4

    const int bz    = blockIdx.z;
    const float* Ab = A + (long long)bz * sAb;
    float*       Cb = C + (long long)bz * sCb;

    const int tid   = threadIdx.x;
    const int lane  = tid & 31;
    const int wave  = tid >> 5;
    const int waveM = wave >> 2;   // 0..1
    const int waveN = wave & 3;    // 0..3
    const int m0    = blockIdx.y * 32;
    const int n0    = blockIdx.x * 128;

    const bool fullM = (m0 + 32 <= M);
    const bool fullN = (n0 + 128 <= N);

    v8f acc0 = {};
    v8f acc1 = {};

    for (int k0 = 0; k0 < K; k0 += 32) {
        // ---- stage A tile (32 x 32) ----
        if (sAk == 1 && fullM) {
            // row-major A: 256 x float4, fully coalesced b128 loads/stores
            int r  = tid >> 3;
            int c4 = (tid & 7) << 2;
            v4f v = *(const v4f*)&Ab[(long long)(m0 + r) * sAm + k0 + c4];
            *(v4f*)&As[r][c4] = v;
        } else if (sAm == 1 && fullM) {
            // 1x1-conv layout: m contiguous in memory -> coalesce over m
            for (int i = tid; i < 32 * 32; i += 256) {
                int r = i & 31, c = i >> 5;
                As[r][c] = Ab[(long long)(m0 + r) + (long long)(k0 + c) * sAk];
            }
        } else {
            for (int i = tid; i < 32 * 32; i += 256) {
                int r = i & 31, c = i >> 5;
                int gm = m0 + r;
                As[r][c] = (gm < M)
                    ? Ab[(long long)gm * sAm + (long long)(k0 + c) * sAk] : 0.f;
            }
        }
        // ---- stage B tile (32 x 128) ----
        if (fullN) {
            #pragma unroll
            for (int j = 0; j < 4; ++j) {
                int idx = tid + j * 256;
                int r   = idx >> 5;
                int c4  = (idx & 31) << 2;
                v4f v = *(const v4f*)&Bm[(long long)(k0 + r) * sBk + n0 + c4];
                *(v4f*)&Bs[r][c4] = v;
            }
        } else {
            for (int i = tid; i < 32 * 128; i += 256) {
                int r = i >> 7, c = i & 127;
                int gn = n0 + c;
                Bs[r][c] = (gn < N) ? Bm[(long long)(k0 + r) * sBk + gn] : 0.f;
            }
        }
        // prefetch next B chunk into cache (global_prefetch_b8)
        if (k0 + 32 < K) {
            int pr = tid >> 3;
            int pc = (tid & 7) << 4;
            if (n0 + pc < N)
                __builtin_prefetch(&Bm[(long long)(k0 + 32 + pr) * sBk + n0 + pc], 0, 0);
        }
        __syncthreads();

        // WMMA fragment lane mapping (ISA 7.12.2, 32-bit A 16x4 / C 16x16):
        // lanes 0-15: M=lane, K pair {0,1}; lanes 16-31: M=lane-16, K pair {2,3}
        const int mm  = (lane & 15) + waveM * 16;
        const int nnA = waveN * 32 + (lane & 15);
        const int nnB = nnA + 16;
        const int kp  = (lane >> 4) << 1;
        #pragma unroll
        for (int kt = 0; kt < 32; kt += 4) {
            v2f af, bf0, bf1;
            af.x  = As[mm][kt + kp];
            af.y  = As[mm][kt + kp + 1];
            bf0.x = Bs[kt + kp][nnA];
            bf0.y = Bs[kt + kp + 1][nnA];
            bf1.x = Bs[kt + kp][nnB];
            bf1.y = Bs[kt + kp + 1][nnB];
            acc0 = __builtin_amdgcn_wmma_f32_16x16x4_f32(
                false, af, false, bf0, (short)0, acc0, false, false);
            acc1 = __builtin_amdgcn_wmma_f32_16x16x4_f32(
                false, af, false, bf1, (short)0, acc1, false, false);
        }
        __syncthreads();
    }

    // C layout: VGPR j -> row (lane>>4)*8 + j, col lane&15
    const int col0    = n0 + waveN * 32 + (lane & 15);
    const int col1    = col0 + 16;
    const int rowBase = m0 + waveM * 16 + (lane >> 4) * 8;
    if (fullM && fullN) {
        const float bv0 = bias ? bias[col0] : 0.f;
        const float bv1 = bias ? bias[col1] : 0.f;
        #pragma unroll
        for (int j = 0; j < 8; ++j) {
            long long r = rowBase + j;
            float v0 = acc0[j] + bv0;
            float v1 = acc1[j] + bv1;
            if (act == 1) { v0 = fmaxf(v0, 0.f); v1 = fmaxf(v1, 0.f); }
            Cb[r * sCm + col0] = v0;
            Cb[r * sCm + col1] = v1;
        }
    } else {
        const float bv0 = (col0 < N && bias) ? bias[col0] : 0.f;
        const float bv1 = (col1 < N && bias) ? bias[col1] : 0.f;
        #pragma unroll
        for (int j = 0; j < 8; ++j) {
            int r = rowBase + j;
            if (r < M) {
                if (col0 < N) {
                    float v = acc0[j] + bv0;
                    if (act == 1) v = fmaxf(v, 0.f);
                    Cb[(long long)r * sCm + col0] = v;
                }
                if (col1 < N) {
                    float v = acc1[j] + bv1;
                    if (act == 1) v = fmaxf(v, 0.f);
                    Cb[(long long)r * sCm + col1] = v;
                }
            }
        }
    }
}

// ---------------------------------------------------------------------------
// GroupNorm over (8 channels x HW) per (group, batch), applied in (B,S,D) layout.
// grid: (32 groups, B). One launch per level.
// ---------------------------------------------------------------------------
__global__ __launch_bounds__(256) void groupnorm_kernel(
    float* __restrict__ src, const float* __restrict__ gw,
    const float* __restrict__ gb, int HW, int start)
{
    const int g = blockIdx.x;
    const int b = blockIdx.y;
    const int n = HW * 8;
    const long long base = ((long long)b * S_TOT + start) * D_;

    float s = 0.f, ss = 0.f;
    for (int e = threadIdx.x; e < n; e += 256) {
        int sp = e >> 3, cc = e & 7;
        float v = src[base + (long long)sp * D_ + g * 8 + cc];
        s += v; ss += v * v;
    }
    __shared__ float rs[256], rss[256];
    rs[threadIdx.x] = s; rss[threadIdx.x] = ss;
    __syncthreads();
    for (int o = 128; o > 0; o >>= 1) {
        if (threadIdx.x < o) { rs[threadIdx.x] += rs[threadIdx.x + o];
                               rss[threadIdx.x] += rss[threadIdx.x + o]; }
        __syncthreads();
    }
    const float mu  = rs[0] / n;
    const float var = rss[0] / n - mu * mu;
    const float inv = rsqrtf(var + 1e-5f);
    for (int e = threadIdx.x; e < n; e += 256) {
        int sp = e >> 3, cc = e & 7;
        int ch = g * 8 + cc;
        long long idx = base + (long long)sp * D_ + ch;
        src[idx] = (src[idx] - mu) * inv * gw[ch] + gb[ch];
    }
}

// query_pos / tgt from broadcast query_embed
__global__ __launch_bounds__(256) void query_init(
    const float* __restrict__ qe, float* __restrict__ qpos, float* __restrict__ tgt)
{
    int i = blockIdx.x * 256 + threadIdx.x;
    if (i >= B_ * NQ_ * D_) return;
    int c = i % D_;
    int q = (i / D_) % NQ_;
    qpos[i] = qe[q * 2 * D_ + c];
    tgt[i]  = qe[q * 2 * D_ + D_ + c];
}

__global__ __launch_bounds__(256) void add_kernel(
    const float* __restrict__ a, const float* __restrict__ b,
    float* __restrict__ o, int n)
{
    int i = blockIdx.x * 256 + threadIdx.x;
    if (i < n) o[i] = a[i] + b[i];
}

// tgt = LayerNorm(tgt + t2) * w + b     (one 256-thread block per row)
__global__ __launch_bounds__(256) void ln_residual(
    float* __restrict__ tgt, const float* __restrict__ t2,
    const float* __restrict__ w, const float* __restrict__ bia)
{
    const int row = blockIdx.x;
    const int c = threadIdx.x;
    float x = tgt[(long long)row * D_ + c] + t2[(long long)row * D_ + c];
    __shared__ float rs[256], rss[256];
    rs[c] = x; rss[c] = x * x;
    __syncthreads();
    for (int o = 128; o > 0; o >>= 1) {
        if (c < o) { rs[c] += rs[c + o]; rss[c] += rss[c + o]; }
        __syncthreads();
    }
    const float mu  = rs[0] / D_;
    const float var = rss[0] / D_ - mu * mu;
    const float inv = rsqrtf(var + 1e-5f);
    tgt[(long long)row * D_ + c] = (x - mu) * inv * w[c] + bia[c];
}

// tiny per-(b,h) self-attention over 42 queries; qkv rows are [q|k|v] of 768
__global__ __launch_bounds__(256) void mha_kernel(
    const float* __restrict__ qkv, float* __restrict__ out)
{
    const int b = blockIdx.x >> 3;
    const int h = blockIdx.x & 7;
    __shared__ float q[NQ_][DH_], k[NQ_][DH_], v[NQ_][DH_], sc[NQ_][NQ_];
    const int tid = threadIdx.x;
    for (int i = tid; i < NQ_ * DH_; i += 256) {
        int qi = i >> 5, d = i & 31;
        long long base = ((long long)(b * NQ_ + qi)) * (3 * D_) + h * DH_ + d;
        q[qi][d] = qkv[base];
        k[qi][d] = qkv[base + D_];
        v[qi][d] = qkv[base + 2 * D_];
    }
    __syncthreads();
    const float scale = 0.1767766952966369f;  // 1/sqrt(32)
    for (int i = tid; i < NQ_ * NQ_; i += 256) {
        int qi = i / NQ_, kj = i % NQ_;
        float s = 0.f;
        #pragma unroll
        for (int d = 0; d < DH_; ++d) s += q[qi][d] * k[kj][d];
        sc[qi][kj] = s * scale;
    }
    __syncthreads();
    if (tid < NQ_) {
        float mx = -1e30f;
        for (int j = 0; j < NQ_; ++j) mx = fmaxf(mx, sc[tid][j]);
        float sum = 0.f;
        for (int j = 0; j < NQ_; ++j) { float e = __expf(sc[tid][j] - mx); sc[tid][j] = e; sum += e; }
        float r = 1.f / sum;
        for (int j = 0; j < NQ_; ++j) sc[tid][j] *= r;
    }
    __syncthreads();
    for (int i = tid; i < NQ_ * DH_; i += 256) {
        int qi = i >> 5, d = i & 31;
        float s = 0.f;
        for (int j = 0; j < NQ_; ++j) s += sc[qi][j] * v[j][d];
        out[((long long)(b * NQ_ + qi)) * D_ + h * DH_ + d] = s;
    }
}

// softmax over the 16 (level,point) weights per (b,q,h)
__global__ __launch_bounds__(256) void softmax16(float* __restrict__ aw)
{
    int i = blockIdx.x * 256 + threadIdx.x;
    if (i >= B_ * NQ_ * NH_) return;
    int bq = i >> 3, h = i & 7;
    float* p = aw + (long long)bq * (NH_ * 16) + h * 16;
    float mx = -1e30f;
    #pragma unroll
    for (int j = 0; j < 16; ++j) mx = fmaxf(mx, p[j]);
    float sum = 0.f;
    #pragma unroll
    for (int j = 0; j < 16; ++j) { float e = __expf(p[j] - mx); p[j] = e; sum += e; }
    float r = 1.f / sum;
    #pragma unroll
    for (int j = 0; j < 16; ++j) p[j] *= r;
}

// multi-scale deformable attention sampling. grid = B*NQ, block(32 d x 8 h)
__global__ __launch_bounds__(256) void ms_deform_kernel(
    const float* __restrict__ value, const float* __restrict__ ref,
    const float* __restrict__ offs, const float* __restrict__ aw,
    float* __restrict__ out)
{
    const int bq = blockIdx.x;
    const int b  = bq / NQ_;
    const int tid = threadIdx.x;
    const int d = tid & 31;
    const int h = tid >> 5;
    const float rx = ref[bq * 2 + 0];
    const float ry = ref[bq * 2 + 1];
    const float* offrow = offs + (long long)bq * D_ + h * 32;       // (lvl,p,2) per head
    const float* awrow  = aw + (long long)bq * (NH_ * 16) + h * 16;

    const int Hs[4]     = {64, 32, 16, 8};
    const int starts[4] = {0, 4096, 5120, 5376};

    float acc = 0.f;
    #pragma unroll
    for (int lvl = 0; lvl < 4; ++lvl) {
        const int H = Hs[lvl], W = Hs[lvl];
        const int start = starts[lvl];
        const float invW = 1.f / (float)W, invH = 1.f / (float)H;
        #pragma unroll
        for (int p = 0; p < 4; ++p) {
            float ox = offrow[lvl * 8 + p * 2 + 0];
            float oy = offrow[lvl * 8 + p * 2 + 1];
            float x = (rx + ox * invW) * W - 0.5f;
            float y = (ry + oy * invH) * H - 0.5f;
            float x0f = floorf(x), y0f = floorf(y);
            float dx = x - x0f, dy = y - y0f;
            int x0 = (int)x0f, y0 = (int)y0f;
            float s = 0.f;
            #pragma unroll
            for (int ci = 0; ci < 4; ++ci) {
                int xi = x0 + (ci & 1);
                int yi = y0 + (ci >> 1);
                float cw = ((ci & 1) ? dx : 1.f - dx) * ((ci >> 1) ? dy : 1.f - dy);
                bool valid = (xi >= 0) && (xi < W) && (yi >= 0) && (yi < H);
                int xc = min(max(xi, 0), W - 1);
                int yc = min(max(yi, 0), H - 1);
                float val = value[((long long)(b * S_TOT + start + yc * W + xc)) * D_ + h * DH_ + d];
                s += valid ? cw * val : 0.f;
            }
            acc += awrow[lvl * 4 + p] * s;
        }
    }
    out[(long long)bq * D_ + h * DH_ + d] = acc;
}

// ref = sigmoid(delta [+ inv_sigmoid(ref)]); optionally write to output slice
__global__ __launch_bounds__(256) void ref_update(
    const float* __restrict__ delta, float* __restrict__ ref,
    float* __restrict__ outslice, int init)
{
    int i = blockIdx.x * 256 + threadIdx.x;
    if (i >= B_ * NQ_ * 2) return;
    float dv = delta[i];
    float v;
    if (init) {
        v = 1.f / (1.f + __expf(-dv));
    } else {
        float r = fminf(fmaxf(ref[i], 0.f), 1.f);
        float a = __logf(fmaxf(r, 1e-5f)) - __logf(fmaxf(1.f - r, 1e-5f));
        v = 1.f / (1.f + __expf(-(dv + a)));
    }
    ref[i] = v;
    if (outslice) outslice[i] = v;
}

// ---------------------------------------------------------------------------
// Host-side orchestration
// ---------------------------------------------------------------------------
static inline void launch_gemm(hipStream_t s,
    const float* A, long long sAb, int sAm, int sAk,
    const float* Bm, int sBk, const float* bias,
    float* C, long long sCb, int sCm,
    int M, int N, int K, int act, int batches)
{
    dim3 grid((N + 127) / 128, (M + 31) / 32, batches);
    gemm_wmma<<<grid, 256, 0, s>>>(A, sAb, sAm, sAk, Bm, sBk, bias, C, sCb, sCm, M, N, K, act);
}

extern "C" void kernel_launch(void* const* d_in, const int* in_sizes, int n_in,
                              void* d_out, int out_size, void* d_ws, size_t ws_size,
                              hipStream_t stream) {
    const float* feat[4]   = {(const float*)d_in[0], (const float*)d_in[1],
                              (const float*)d_in[2], (const float*)d_in[3]};
    const float* proj_w    = (const float*)d_in[4];
    const float* proj_b    = (const float*)d_in[5];
    const float* gn_w      = (const float*)d_in[6];
    const float* gn_b      = (const float*)d_in[7];
    const float* query_emb = (const float*)d_in[8];
    const float* bbox_w1   = (const float*)d_in[9];
    const float* bbox_b1   = (const float*)d_in[10];
    const float* bbox_w2   = (const float*)d_in[11];
    const float* bbox_b2   = (const float*)d_in[12];
    const float* bbox_w3   = (const float*)d_in[13];
    const float* bbox_b3   = (const float*)d_in[14];
    const float* sa_in_w   = (const float*)d_in[15];
    const float* sa_in_b   = (const float*)d_in[16];
    const float* sa_out_w  = (const float*)d_in[17];
    const float* sa_out_b  = (const float*)d_in[18];
    const float* n1_w      = (const float*)d_in[19];
    const float* n1_b      = (const float*)d_in[20];
    const float* n2_w      = (const float*)d_in[21];
    const float* n2_b      = (const float*)d_in[22];
    const float* n3_w      = (const float*)d_in[23];
    const float* n3_b      = (const float*)d_in[24];
    const float* ca_off_w  = (const float*)d_in[25];
    const float* ca_off_b  = (const float*)d_in[26];
    const float* ca_attw_w = (const float*)d_in[27];
    const float* ca_attw_b = (const float*)d_in[28];
    const float* ca_val_w  = (const float*)d_in[29];
    const float* ca_val_b  = (const float*)d_in[30];
    const float* ca_out_w  = (const float*)d_in[31];
    const float* ca_out_b  = (const float*)d_in[32];
    const float* ffn_w1    = (const float*)d_in[33];
    const float* ffn_b1    = (const float*)d_in[34];
    const float* ffn_w2    = (const float*)d_in[35];
    const float* ffn_b2    = (const float*)d_in[36];

    // workspace partition (all fp32)
    float* p = (float*)d_ws;
    float* src    = p; p += (size_t)B_ * S_TOT * D_;
    float* value  = p; p += (size_t)B_ * S_TOT * D_;
    float* qpos   = p; p += (size_t)B_ * NQ_ * D_;
    float* tgt    = p; p += (size_t)B_ * NQ_ * D_;
    float* qin    = p; p += (size_t)B_ * NQ_ * D_;
    float* qkv    = p; p += (size_t)B_ * NQ_ * 3 * D_;
    float* attn   = p; p += (size_t)B_ * NQ_ * D_;
    float* t2     = p; p += (size_t)B_ * NQ_ * D_;
    float* h1     = p; p += (size_t)B_ * NQ_ * D_;
    float* h2     = p; p += (size_t)B_ * NQ_ * D_;
    float* delta  = p; p += (size_t)B_ * NQ_ * 2;
    float* hidden = p; p += (size_t)B_ * NQ_ * FF_;
    float* offs   = p; p += (size_t)B_ * NQ_ * D_;
    float* attw   = p; p += (size_t)B_ * NQ_ * NH_ * 16;
    float* samp   = p; p += (size_t)B_ * NQ_ * D_;
    float* ref    = p; p += (size_t)B_ * NQ_ * 2;

    const int Hs[4]     = {64, 32, 16, 8};
    const int starts[4] = {0, 4096, 5120, 5376};
    const int MQ = B_ * NQ_;           // 1344
    const int NE = MQ * D_;            // elementwise count

    // ---- input projection (1x1 conv as strided GEMM) + GroupNorm ----
    for (int l = 0; l < 4; ++l) {
        const int HW = Hs[l] * Hs[l];
        launch_gemm(stream,
            feat[l], (long long)D_ * HW, /*sAm=*/1, /*sAk=*/HW,
            proj_w + (size_t)l * D_ * D_, D_, proj_b + (size_t)l * D_,
            src + (long long)starts[l] * D_, (long long)S_TOT * D_, D_,
            HW, D_, D_, 0, B_);
        groupnorm_kernel<<<dim3(32, B_), 256, 0, stream>>>(
            src, gn_w + (size_t)l * D_, gn_b + (size_t)l * D_, HW, starts[l]);
    }

    // ---- queries ----
    query_init<<<(NE + 255) / 256, 256, 0, stream>>>(query_emb, qpos, tgt);

    // initial ref = sigmoid(bbox_mlp(query_pos))
    launch_gemm(stream, qpos, 0, D_, 1, bbox_w1, D_, bbox_b1, h1, 0, D_, MQ, D_, D_, 1, 1);
    launch_gemm(stream, h1,   0, D_, 1, bbox_w2, D_, bbox_b2, h2, 0, D_, MQ, D_, D_, 1, 1);
    launch_gemm(stream, h2,   0, D_, 1, bbox_w3, 2,  bbox_b3, delta, 0, 2, MQ, 2, D_, 0, 1);
    ref_update<<<(MQ * 2 + 255) / 256, 256, 0, stream>>>(delta, ref, nullptr, 1);

    // ---- decoder layers ----
    for (int i = 0; i < NLAYERS_; ++i) {
        const float* saw = sa_in_w + (size_t)i * D_ * 3 * D_;
        const float* sab = sa_in_b + (size_t)i * 3 * D_;

        // self-attention
        add_kernel<<<(NE + 255) / 256, 256, 0, stream>>>(tgt, qpos, qin, NE);
        launch_gemm(stream, qin, 0, D_, 1, saw, 3 * D_, sab,
                    qkv, 0, 3 * D_, MQ, 2 * D_, D_, 0, 1);                 // q|k
        launch_gemm(stream, tgt, 0, D_, 1, saw + 2 * D_, 3 * D_, sab + 2 * D_,
                    qkv + 2 * D_, 0, 3 * D_, MQ, D_, D_, 0, 1);            // v
        mha_kernel<<<B_ * NH_, 256, 0, stream>>>(qkv, attn);
        launch_gemm(stream, attn, 0, D_, 1, sa_out_w + (size_t)i * D_ * D_, D_,
                    sa_out_b + (size_t)i * D_, t2, 0, D_, MQ, D_, D_, 0, 1);
        ln_residual<<<MQ, 256, 0, stream>>>(tgt, t2, n1_w + (size_t)i * D_, n1_b + (size_t)i * D_);

        // cross-attention (multi-scale deformable)
        add_kernel<<<(NE + 255) / 256, 256, 0, stream>>>(tgt, qpos, qin, NE);
        launch_gemm(stream, src, 0, D_, 1, ca_val_w + (size_t)i * D_ * D_, D_,
                    ca_val_b + (size_t)i * D_, value, 0, D_,
                    B_ * S_TOT, D_, D_, 0, 1);
        launch_gemm(stream, qin, 0, D_, 1, ca_off_w + (size_t)i * D_ * 256, 256,
                    ca_off_b + (size_t)i * 256, offs, 0, 256, MQ, 256, D_, 0, 1);
        launch_gemm(stream, qin, 0, D_, 1, ca_attw_w + (size_t)i * D_ * 128, 128,
                    ca_attw_b + (size_t)i * 128, attw, 0, 128, MQ, 128, D_, 0, 1);
        softmax16<<<(MQ * NH_ + 255) / 256, 256, 0, stream>>>(attw);
        ms_deform_kernel<<<MQ, 256, 0, stream>>>(value, ref, offs, attw, samp);
        launch_gemm(stream, samp, 0, D_, 1, ca_out_w + (size_t)i * D_ * D_, D_,
                    ca_out_b + (size_t)i * D_, t2, 0, D_, MQ, D_, D_, 0, 1);
        ln_residual<<<MQ, 256, 0, stream>>>(tgt, t2, n2_w + (size_t)i * D_, n2_b + (size_t)i * D_);

        // FFN
        launch_gemm(stream, tgt, 0, D_, 1, ffn_w1 + (size_t)i * D_ * FF_, FF_,
                    ffn_b1 + (size_t)i * FF_, hidden, 0, FF_, MQ, FF_, D_, 1, 1);
        launch_gemm(stream, hidden, 0, FF_, 1, ffn_w2 + (size_t)i * FF_ * D_, D_,
                    ffn_b2 + (size_t)i * D_, t2, 0, D_, MQ, D_, FF_, 0, 1);
        ln_residual<<<MQ, 256, 0, stream>>>(tgt, t2, n3_w + (size_t)i * D_, n3_b + (size_t)i * D_);

        // bbox refinement
        launch_gemm(stream, tgt, 0, D_, 1, bbox_w1, D_, bbox_b1, h1, 0, D_, MQ, D_, D_, 1, 1);
        launch_gemm(stream, h1,  0, D_, 1, bbox_w2, D_, bbox_b2, h2, 0, D_, MQ, D_, D_, 1, 1);
        launch_gemm(stream, h2,  0, D_, 1, bbox_w3, 2,  bbox_b3, delta, 0, 2, MQ, 2, D_, 0, 1);
        ref_update<<<(MQ * 2 + 255) / 256, 256, 0, stream>>>(
            delta, ref, (float*)d_out + (size_t)i * MQ * 2, 0);
    }
}